// KDEODF_3178275799516
// MI455X (gfx1250) — compile-verified
//
#include <hip/hip_runtime.h>

// CDNA5 / gfx1250 KDE over rotations:
//   out[m] = (KAPPA+1)/N_X * sum_n min(|qy[m]·qx[n]|, 1-1e-7)^(2*KAPPA)
// K=4 quaternion dots via V_WMMA_F32_16X16X4_F32; c^100 via packed
// v_pk_mul_f32 / v_pk_fma_f32 (minimal chain: c^100 = (c^2)^48 * (c^2)^2,
// last multiply fused into the accumulate FMA).

typedef __attribute__((ext_vector_type(2))) float v2f;
typedef __attribute__((ext_vector_type(8))) float v8f;

#define KAPPA_P1 51.0f          // KAPPA + 1
#define CLAMP2   0.9999998f     // (1 - 1e-7)^2

// ---- packed f32 math (VOP3P) -------------------------------------------
__device__ __forceinline__ v2f pk_mul(v2f a, v2f b) {
  v2f d;
  asm("v_pk_mul_f32 %0, %1, %2" : "=v"(d) : "v"(a), "v"(b));
  return d;
}
__device__ __forceinline__ v2f pk_fma(v2f a, v2f b, v2f c) {
  v2f d;
  asm("v_pk_fma_f32 %0, %1, %2, %3" : "=v"(d) : "v"(a), "v"(b), "v"(c));
  return d;
}
// raw v_min_num_f32 (avoids the compiler's NaN-canonicalize v_max before min)
__device__ __forceinline__ float vmin(float a, float b) {
  float d;
  asm("v_min_num_f32 %0, %1, %2" : "=v"(d) : "v"(a), "v"(b));
  return d;
}

// acc += min(v*v, CLAMP2)^50   (== c^100), on a pair of elements
__device__ __forceinline__ void accum_pair(v2f v, v2f& acc, float clamp2) {
  v2f s = pk_mul(v, v);                 // c^2  (>=0, so no abs needed)
  s[0] = vmin(s[0], clamp2);
  s[1] = vmin(s[1], clamp2);
  v2f p4  = pk_mul(s,   s);             // c^4   = s^2
  v2f p8  = pk_mul(p4,  p4);            // c^8
  v2f p16 = pk_mul(p8,  p8);            // c^16
  v2f p32 = pk_mul(p16, p16);           // c^32
  v2f p64 = pk_mul(p32, p32);           // c^64
  v2f p96 = pk_mul(p64, p32);           // c^96  = s^48
  acc = pk_fma(p96, p4, acc);           // acc += s^48 * s^2 = c^100
}

// ---- main kernel --------------------------------------------------------
// Block = 256 threads = 8 waves. Each wave owns 16 Y queries (B tile in
// registers) and loops over all X tiles (A prefetched from LDS; LDS is
// padded by one tile so the prefetch needs no wrap/guard logic).
__global__ __launch_bounds__(256) void kde_odf_wmma(
    const float* __restrict__ X,   // [n_x, 3] MRP
    const float* __restrict__ Y,   // [n_y, 3] MRP
    float* __restrict__ out,       // [n_y]
    int n_x, int n_y) {
  extern __shared__ float qx[];    // (n_x + 16) rows * 4 floats, A layout

  const int t = threadIdx.x;

  // --- stage X quaternions into LDS (coalesced global, conflict-free ds) ---
  for (int row = t; row < n_x; row += 256) {
    float rx = X[row * 3 + 0];
    float ry = X[row * 3 + 1];
    float rz = X[row * 3 + 2];
    float rr  = rx * rx + ry * ry + rz * rz;
    float inv = 1.0f / (1.0f + rr);
    qx[row * 4 + 0] = (1.0f - rr) * inv;   // w
    qx[row * 4 + 1] = 2.0f * rx * inv;     // x
    qx[row * 4 + 2] = 2.0f * ry * inv;     // y
    qx[row * 4 + 3] = 2.0f * rz * inv;     // z
  }
  __syncthreads();

  const int lane = t & 31;
  const int wave = t >> 5;
  const int half = lane >> 4;      // 0: K={0,1}/rows 0..7 ; 1: K={2,3}/rows 8..15
  const int l16  = lane & 15;
  const int yrow = blockIdx.x * 128 + wave * 16 + l16;

  // --- B matrix (4 x 16, qy tile), fixed per wave -------------------------
  // B VGPR0 = K0 (lanes 0-15) / K2 (lanes 16-31); VGPR1 = K1 / K3.
  float ry0 = Y[yrow * 3 + 0];
  float ry1 = Y[yrow * 3 + 1];
  float ry2 = Y[yrow * 3 + 2];
  float brr  = ry0 * ry0 + ry1 * ry1 + ry2 * ry2;
  float binv = 1.0f / (1.0f + brr);
  float qw = (1.0f - brr) * binv;
  float q1 = 2.0f * ry0 * binv;
  float q2 = 2.0f * ry1 * binv;
  float q3 = 2.0f * ry2 * binv;
  v2f b;
  b[0] = half ? q2 : qw;
  b[1] = half ? q3 : q1;

  // --- loop over X tiles, software-pipelined unguarded A prefetch ----------
  const float clamp2 = CLAMP2;
  v2f acc = {0.0f, 0.0f};
  const int ntiles = n_x >> 4;
  const int abase  = l16 * 4 + half * 2;     // float offset within a tile block

  v2f a = *(const v2f*)(&qx[abase]);         // prologue: tile 0's A pair
#pragma unroll 2
  for (int tile = 0; tile < ntiles; ++tile) {
    v8f c = {};
    v8f d = __builtin_amdgcn_wmma_f32_16x16x4_f32(
        /*neg_a=*/false, a, /*neg_b=*/false, b,
        /*c_mod=*/(short)0, c, /*reuse_a=*/false, /*reuse_b=*/false);
    // unguarded prefetch of next tile's A (last one lands in the LDS pad)
    a = *(const v2f*)(&qx[(tile + 1) * 64 + abase]);
    // D: lane = Y column, 8 VGPRs = 8 X rows (this half). Reduce locally.
    v2f v0 = {d[0], d[1]};
    v2f v1 = {d[2], d[3]};
    v2f v2 = {d[4], d[5]};
    v2f v3 = {d[6], d[7]};
    accum_pair(v0, acc, clamp2);
    accum_pair(v1, acc, clamp2);
    accum_pair(v2, acc, clamp2);
    accum_pair(v3, acc, clamp2);
  }

  // --- finish reduction: pair-sum, then merge the two 8-row halves ---------
  float colsum = acc[0] + acc[1];
  colsum += __shfl_xor(colsum, 16, 32);
  if (half == 0) {
    out[yrow] = colsum * (KAPPA_P1 / (float)n_x);
  }
}

// ---- launcher -----------------------------------------------------------
extern "C" void kernel_launch(void* const* d_in, const int* in_sizes, int n_in,
                              void* d_out, int out_size, void* d_ws, size_t ws_size,
                              hipStream_t stream) {
  const float* X = (const float*)d_in[0];
  const float* Y = (const float*)d_in[1];
  float* out = (float*)d_out;
  const int n_x = in_sizes[0] / 3;   // 4096
  const int n_y = in_sizes[1] / 3;   // 65536

  const int blocks = n_y / 128;      // 128 Y queries / block
  // qx quaternions + one padding tile (16 rows) for the unguarded prefetch
  const size_t shmem = ((size_t)n_x + 16) * 4 * sizeof(float);
  hipLaunchKernelGGL(kde_odf_wmma, dim3(blocks), dim3(256), shmem, stream,
                     X, Y, out, n_x, n_y);
}